// SpectralAttentionModule_2508260901050
// MI455X (gfx1250) — compile-verified
//
#include <hip/hip_runtime.h>
#include <math.h>

// ---------------------------------------------------------------------------
// Spectral attention: rfft -> QKV -> |QK^T| softmax -> PV -> irfft
// Precision: fp32 end-to-end, matrix cores via V_WMMA_F32_16X16X4_F32.
// ---------------------------------------------------------------------------

typedef float v2f __attribute__((ext_vector_type(2)));
typedef float v8f __attribute__((ext_vector_type(8)));

#define BGN   160          // B*G = 4*40
#define CDIM  13
#define TDIM  1024
#define FDIM  513          // rfft bins
#define FP    528          // FDIM padded to multiple of 16 (33 tiles)
#define CP    16           // channel dim padded to 16
#define NT    33           // FP/16
#define WAVES 8

// WMMA f32 16x16x4 accumulate wrapper (8-arg form: neg_a, A, neg_b, B, c_mod, C, reuse_a, reuse_b)
__device__ __forceinline__ v8f wmma_f32_k4(v2f a, v2f b, v8f c) {
    return __builtin_amdgcn_wmma_f32_16x16x4_f32(false, a, false, b, (short)0, c, false, false);
}

// ---------------------------------------------------------------------------
// In-LDS radix-2 DIT FFT over 1024 complex points (input pre-bit-reversed).
// sign = -1: forward (e^{-i2pi kn/N}); sign = +1: inverse (unscaled).
// ---------------------------------------------------------------------------
__device__ void fft1024(float2* a, int tid, float sign) {
    for (int s = 1; s <= 10; ++s) {
        const int half = 1 << (s - 1);
        const int len  = 1 << s;
        const float w0 = sign * 6.28318530717958647692f / (float)len;
        for (int b = tid; b < 512; b += 256) {
            const int j    = b & (half - 1);
            const int base = ((b >> (s - 1)) << s) + j;
            float sn, cs;
            __sincosf(w0 * (float)j, &sn, &cs);
            float2 u = a[base];
            float2 t = a[base + half];
            float tr = t.x * cs - t.y * sn;
            float ti = t.x * sn + t.y * cs;
            a[base]        = make_float2(u.x + tr, u.y + ti);
            a[base + half] = make_float2(u.x - tr, u.y - ti);
        }
        __syncthreads();
    }
}

__device__ __forceinline__ int rev10(int i) { return (int)(__brev((unsigned)i) >> 22); }

// ---------------------------------------------------------------------------
// Kernel 1: rfft over time for each (bg, c) row; write transposed xt[bg][f][c]
// ---------------------------------------------------------------------------
__global__ __launch_bounds__(256) void rfft_kernel(const float* __restrict__ x,
                                                   float2* __restrict__ xws) {
    __shared__ float2 a[TDIM];
    const int row = blockIdx.x;            // bg*13 + c, 2080 rows
    const int bg  = row / CDIM;
    const int c   = row % CDIM;
    const int tid = threadIdx.x;

    const float* src = x + (size_t)row * TDIM;
    for (int t = tid; t < TDIM; t += 256)
        a[rev10(t)] = make_float2(src[t], 0.0f);
    __syncthreads();

    fft1024(a, tid, -1.0f);

    for (int f = tid; f < FDIM; f += 256)
        xws[((size_t)bg * FP + f) * CP + c] = a[f];
}

// ---------------------------------------------------------------------------
// Kernel 2: QKV projections. q/k/v[bg][f][j] (complex), f padded to FP and
// j padded to CP with zeros so the WMMA tiles need no masking.
// ---------------------------------------------------------------------------
__global__ __launch_bounds__(256) void qkv_kernel(const float2* __restrict__ xws,
                                                  const float* __restrict__ wq,
                                                  const float* __restrict__ wk,
                                                  const float* __restrict__ wv,
                                                  float2* __restrict__ q,
                                                  float2* __restrict__ k,
                                                  float2* __restrict__ v) {
    const int gid = blockIdx.x * 256 + threadIdx.x;      // BGN*FP*CP threads
    const int j   = gid & (CP - 1);
    const int fb  = gid >> 4;
    const int f   = fb % FP;
    const int bg  = fb / FP;
    if (bg >= BGN) return;

    float2 zq = make_float2(0.f, 0.f);
    float2 zk = make_float2(0.f, 0.f);
    float2 zv = make_float2(0.f, 0.f);
    if (f < FDIM && j < CDIM) {
        const float2* xr = xws + ((size_t)bg * FP + f) * CP;
#pragma unroll
        for (int c = 0; c < CDIM; ++c) {
            float2 xv = xr[c];
            float aq = wq[c * CDIM + j];
            float ak = wk[c * CDIM + j];
            float av = wv[c * CDIM + j];
            zq.x += xv.x * aq; zq.y += xv.y * aq;
            zk.x += xv.x * ak; zk.y += xv.y * ak;
            zv.x += xv.x * av; zv.y += xv.y * av;
        }
    }
    const size_t o = ((size_t)bg * FP + f) * CP + j;
    q[o] = zq; k[o] = zk; v[o] = zv;
}

// ---------------------------------------------------------------------------
// Kernel 3: per (bg, 16-row strip): scores via WMMA, softmax, PV via WMMA.
// ---------------------------------------------------------------------------
__global__ __launch_bounds__(256) void attn_kernel(const float2* __restrict__ qws,
                                                   const float2* __restrict__ kws,
                                                   const float2* __restrict__ vws,
                                                   float2* __restrict__ ows) {
    __shared__ float S[16][FP + 8];          // score/prob strip, padded stride
    __shared__ float redRe[WAVES][16][16];
    __shared__ float redIm[WAVES][16][16];
    __shared__ float rtmp[16][16];
    __shared__ float rowmax[16];
    __shared__ float rowsum[16];

    const int tid  = threadIdx.x;
    const int wave = tid >> 5;
    const int lane = tid & 31;
    const int i0   = blockIdx.x * 16;        // row-strip base (33 strips)
    const int bg   = blockIdx.y;

    const int m  = lane & 15;                // A row / B col / C col index
    const int hi = lane >> 4;                // lane half
    const int kh = hi * 2;                   // K sub-offset within 4-step

    const float2* qb = qws + (size_t)bg * FP * CP;
    const float2* kb = kws + (size_t)bg * FP * CP;
    const float2* vb = vws + (size_t)bg * FP * CP;

    // ---- Phase 1: scores strip S[16][FP] = |Q_strip . K^T| -----------------
    for (int jt = wave; jt < NT; jt += WAVES) {
        const int j0 = jt * 16;
        v8f accRR = {0,0,0,0,0,0,0,0};
        v8f accII = {0,0,0,0,0,0,0,0};
        v8f accRI = {0,0,0,0,0,0,0,0};
        v8f accIR = {0,0,0,0,0,0,0,0};
#pragma unroll
        for (int s = 0; s < 4; ++s) {
            const int c0 = s * 4 + kh;
            float2 q0 = qb[(size_t)(i0 + m) * CP + c0];
            float2 q1 = qb[(size_t)(i0 + m) * CP + c0 + 1];
            float2 k0 = kb[(size_t)(j0 + m) * CP + c0];
            float2 k1 = kb[(size_t)(j0 + m) * CP + c0 + 1];
            v2f aR = {q0.x, q1.x}, aI = {q0.y, q1.y};
            v2f bR = {k0.x, k1.x}, bI = {k0.y, k1.y};
            accRR = wmma_f32_k4(aR, bR, accRR);
            accII = wmma_f32_k4(aI, bI, accII);
            accRI = wmma_f32_k4(aR, bI, accRI);
            accIR = wmma_f32_k4(aI, bR, accIR);
        }
#pragma unroll
        for (int e = 0; e < 8; ++e) {
            float sre = accRR[e] - accII[e];
            float sim = accRI[e] + accIR[e];
            const int r   = e + hi * 8;
            const int col = j0 + m;
            S[r][col] = (col < FDIM) ? sqrtf(sre * sre + sim * sim) : -3.0e38f;
        }
    }
    __syncthreads();

    // ---- Phase 2: softmax across FP columns (pad cols contribute exp->0) ---
    {
        const int r  = tid >> 4;
        const int cl = tid & 15;
        float lmax = -3.0e38f;
        for (int col = cl; col < FP; col += 16) lmax = fmaxf(lmax, S[r][col]);
        rtmp[r][cl] = lmax;
        __syncthreads();
        if (cl == 0) {
            float mx = rtmp[r][0];
#pragma unroll
            for (int i = 1; i < 16; ++i) mx = fmaxf(mx, rtmp[r][i]);
            rowmax[r] = mx;
        }
        __syncthreads();
        const float mx = rowmax[r];
        float lsum = 0.0f;
        for (int col = cl; col < FP; col += 16) {
            float e = __expf(S[r][col] - mx);
            S[r][col] = e;
            lsum += e;
        }
        rtmp[r][cl] = lsum;
        __syncthreads();
        if (cl == 0) {
            float sm = 0.0f;
#pragma unroll
            for (int i = 0; i < 16; ++i) sm += rtmp[r][i];
            rowsum[r] = sm;
        }
        __syncthreads();
        const float inv = 1.0f / rowsum[r];
        for (int col = cl; col < FP; col += 16) S[r][col] *= inv;
    }
    __syncthreads();

    // ---- Phase 3: out_strip = P[16,FP] @ V[FP,CP] (re & im) ----------------
    v8f accRe = {0,0,0,0,0,0,0,0};
    v8f accIm = {0,0,0,0,0,0,0,0};
    for (int kt = wave; kt < NT; kt += WAVES) {
        const int k0 = kt * 16;
#pragma unroll
        for (int s = 0; s < 4; ++s) {
            const int kk = k0 + s * 4 + kh;
            v2f a = {S[m][kk], S[m][kk + 1]};
            float2 v0 = vb[(size_t)kk * CP + m];
            float2 v1 = vb[(size_t)(kk + 1) * CP + m];
            v2f bR = {v0.x, v1.x}, bI = {v0.y, v1.y};
            accRe = wmma_f32_k4(a, bR, accRe);
            accIm = wmma_f32_k4(a, bI, accIm);
        }
    }
#pragma unroll
    for (int e = 0; e < 8; ++e) {
        redRe[wave][e + hi * 8][m] = accRe[e];
        redIm[wave][e + hi * 8][m] = accIm[e];
    }
    __syncthreads();

    // deterministic cross-wave reduction + store out[bg][c][f]
    {
        const int rr = tid >> 4;
        const int cc = tid & 15;
        float sre = 0.0f, sim = 0.0f;
#pragma unroll
        for (int w = 0; w < WAVES; ++w) { sre += redRe[w][rr][cc]; sim += redIm[w][rr][cc]; }
        const int f = i0 + rr;
        if (cc < CDIM && f < FDIM)
            ows[((size_t)bg * CDIM + cc) * FDIM + f] = make_float2(sre, sim);
    }
}

// ---------------------------------------------------------------------------
// Kernel 4: irfft per (bg, c) row -> d_out[B,G,C,T]
// ---------------------------------------------------------------------------
__global__ __launch_bounds__(256) void irfft_kernel(const float2* __restrict__ ows,
                                                    float* __restrict__ out) {
    __shared__ float2 a[TDIM];
    const int row = blockIdx.x;              // bg*13 + c
    const int tid = threadIdx.x;

    const float2* src = ows + (size_t)row * FDIM;
    for (int t = tid; t < TDIM; t += 256) {
        float2 val;
        if (t <= 512) {
            val = src[t];
        } else {
            float2 s = src[TDIM - t];
            val = make_float2(s.x, -s.y);    // Hermitian extension
        }
        a[rev10(t)] = val;
    }
    __syncthreads();

    fft1024(a, tid, +1.0f);

    const float scale = 1.0f / (float)TDIM;
    float* dst = out + (size_t)row * TDIM;
    for (int t = tid; t < TDIM; t += 256)
        dst[t] = a[t].x * scale;             // real part discards Im(Y0/Y512)
}

// ---------------------------------------------------------------------------
extern "C" void kernel_launch(void* const* d_in, const int* in_sizes, int n_in,
                              void* d_out, int out_size, void* d_ws, size_t ws_size,
                              hipStream_t stream) {
    (void)in_sizes; (void)n_in; (void)out_size; (void)ws_size;

    const float* x  = (const float*)d_in[0];   // [4,40,13,1024]
    const float* wq = (const float*)d_in[1];   // [13,13]
    const float* wk = (const float*)d_in[2];
    const float* wv = (const float*)d_in[3];
    float* out = (float*)d_out;                // [4,40,13,1024]

    // workspace layout (float2 elements)
    const size_t nPad = (size_t)BGN * FP * CP;           // padded [bg][f][c]
    float2* xws = (float2*)d_ws;
    float2* qws = xws + nPad;
    float2* kws = qws + nPad;
    float2* vws = kws + nPad;
    float2* ows = vws + nPad;                            // [bg][c][513]

    const int rows = BGN * CDIM;                         // 2080 FFT rows

    rfft_kernel<<<rows, 256, 0, stream>>>(x, xws);

    const int qkvThreads = BGN * FP * CP;
    qkv_kernel<<<(qkvThreads + 255) / 256, 256, 0, stream>>>(xws, wq, wk, wv,
                                                             qws, kws, vws);

    attn_kernel<<<dim3(NT, BGN), 256, 0, stream>>>(qws, kws, vws, ows);

    irfft_kernel<<<rows, 256, 0, stream>>>(ows, out);
}